// HookeanSprings_10230612099765
// MI455X (gfx1250) — compile-verified
//
#include <hip/hip_runtime.h>
#include <stdint.h>

// HookeanSprings energy: e = 0.5 * sum_e k[e] * (|x[i_e]-x[j_e]| - l0[e])^2
//
// Memory-bound streaming reduction. CDNA5 path used: double-buffered
// global_load_async_to_lds_b128 (ASYNCcnt) to pipeline the 128MB int64 index
// stream into LDS while the wave performs L2-resident vertex gathers and VALU.

#define BLOCK 256
#define GRID  2048

__global__ __launch_bounds__(BLOCK) void springs_main_kernel(
    const float* __restrict__ x,        // [V,3]
    const long long* __restrict__ idx,  // [E,2] int64
    const float* __restrict__ l0,       // [E]
    const float* __restrict__ kk,       // [E]
    float* __restrict__ partials,       // [gridDim.x]
    int E, int niters)
{
    // double-buffered index staging: 2 bufs * 256 lanes * 16B = 8KB
    __shared__ long long sIdx[2][2 * BLOCK];
    __shared__ float red[BLOCK];

    const int tid = threadIdx.x;
    const int gid = blockIdx.x * BLOCK + tid;
    const int S   = gridDim.x * BLOCK;

    // Issue async copy of edge (gid + it*S)'s int64 pair into LDS buffer b.
    // Index clamped to stay in-bounds; masked at accumulate time.
    auto issue_async = [&](int it, int b) {
        int e  = gid + it * S;
        int ec = (e < E) ? e : (E - 1);
        uint32_t lds32 = (uint32_t)(uintptr_t)(&sIdx[b][2 * tid]); // low 32b = LDS offset
        uint64_t ga    = (uint64_t)(uintptr_t)(idx + 2 * (long long)ec);
        asm volatile("global_load_async_to_lds_b128 %0, %1, off"
                     :: "v"(lds32), "v"(ga)
                     : "memory");
    };

    float acc = 0.0f;

    issue_async(0, 0);
    for (int it = 0; it < niters; ++it) {
        const int b = it & 1;
        const bool more = (it + 1 < niters);   // uniform across block
        if (more) {
            issue_async(it + 1, b ^ 1);
            // one outstanding async (next tile) allowed; current tile done
            asm volatile("s_wait_asynccnt 0x1" ::: "memory");
        } else {
            asm volatile("s_wait_asynccnt 0x0" ::: "memory");
        }

        int e  = gid + it * S;
        int ec = (e < E) ? e : (E - 1);

        long long i = sIdx[b][2 * tid + 0];    // ds_load (own 16B slot)
        long long j = sIdx[b][2 * tid + 1];

        const float* xi = x + 3 * i;           // L2-resident gathers
        const float* xj = x + 3 * j;
        float dx = xi[0] - xj[0];
        float dy = xi[1] - xj[1];
        float dz = xi[2] - xj[2];

        float q  = fmaf(dx, dx, fmaf(dy, dy, dz * dz));
        float l  = sqrtf(q + 1e-6f);
        float dl = l - l0[ec];
        float w  = (e < E) ? kk[ec] : 0.0f;    // mask tail, keep EXEC full
        acc = fmaf(w * dl, dl, acc);
    }

    // fixed-order block tree reduction (deterministic)
    red[tid] = acc;
    __syncthreads();
    #pragma unroll
    for (int s = BLOCK / 2; s > 0; s >>= 1) {
        if (tid < s) red[tid] += red[tid + s];
        __syncthreads();
    }
    if (tid == 0) partials[blockIdx.x] = red[0];
}

__global__ __launch_bounds__(BLOCK) void springs_reduce_kernel(
    const float* __restrict__ partials, int n, float* __restrict__ out)
{
    __shared__ float red[BLOCK];
    const int tid = threadIdx.x;
    float a = 0.0f;
    for (int i = tid; i < n; i += BLOCK) a += partials[i];  // fixed order
    red[tid] = a;
    __syncthreads();
    #pragma unroll
    for (int s = BLOCK / 2; s > 0; s >>= 1) {
        if (tid < s) red[tid] += red[tid + s];
        __syncthreads();
    }
    if (tid == 0) out[0] = 0.5f * red[0];
}

extern "C" void kernel_launch(void* const* d_in, const int* in_sizes, int n_in,
                              void* d_out, int out_size, void* d_ws, size_t ws_size,
                              hipStream_t stream) {
    const float*     x   = (const float*)d_in[0];      // [V,3] f32
    const long long* idx = (const long long*)d_in[1];  // [E,2] i64
    const float*     l0  = (const float*)d_in[2];      // [E]   f32
    const float*     kk  = (const float*)d_in[3];      // [E]   f32
    float*           out = (float*)d_out;              // scalar f32

    const int E = in_sizes[2];                          // l0 length == E

    int blocks = GRID;
    int maxb = (int)(ws_size / sizeof(float));
    if (blocks > maxb) blocks = maxb > 0 ? maxb : 1;

    float* partials = (float*)d_ws;
    const int S = blocks * BLOCK;
    const int niters = (E + S - 1) / S;

    springs_main_kernel<<<blocks, BLOCK, 0, stream>>>(x, idx, l0, kk, partials, E, niters);
    springs_reduce_kernel<<<1, BLOCK, 0, stream>>>(partials, blocks, out);
}